// LIF_45621142618288
// MI455X (gfx1250) — compile-verified
//
#include <hip/hip_runtime.h>

// LIF spiking-neuron forward pass, T=4 sequential steps over [T, B*N] fp32.
// Memory-bound streaming kernel: 268 MB total traffic -> ~11.5 us floor at
// 23.3 TB/s on MI455X. Strategy: b128 non-temporal loads/stores (single-pass
// stream, don't pollute WGP$/L2), all 4 timestep loads in flight per lane,
// full T-recurrence held in VGPRs. 32-bit indexing so addressing lowers to
// SGPR-base + 32-bit VGPR offset instead of 64-bit VALU address chains.

#define T_STEPS 4
#define THRESH  1.0f
#define TAU     1.0f

typedef float f4 __attribute__((ext_vector_type(4)));

__global__ __launch_bounds__(256) void lif_fwd_kernel(
    const f4* __restrict__ x,   // [T, perT4] in f4 units
    f4* __restrict__ out,       // [T, perT4] in f4 units
    unsigned perT4)             // (B*N)/4 f4-elements per timestep
{
    unsigned idx = blockIdx.x * blockDim.x + threadIdx.x;  // f4 column index
    if (idx >= perT4) return;

    // Issue all 4 timestep loads up-front: 4 outstanding global_load_b128,
    // non-temporal (data is read exactly once).
    f4 xv[T_STEPS];
#pragma unroll
    for (int t = 0; t < T_STEPS; ++t) {
        xv[t] = __builtin_nontemporal_load(x + ((unsigned)t * perT4 + idx));
    }

    // LIF recurrence per vector component, entirely in VGPRs.
    f4 ov[T_STEPS];
#pragma unroll
    for (int c = 0; c < 4; ++c) {
        float mem = 0.0f;
#pragma unroll
        for (int t = 0; t < T_STEPS; ++t) {
            mem = TAU * mem + xv[t][c];
            float h     = (mem - THRESH >= 0.0f) ? 1.0f : 0.0f;  // Heaviside
            float spike = h * THRESH;
            ov[t][c] = spike;
            mem = mem * (1.0f - spike);  // THRESH==1 -> reset to 0 on spike
        }
    }

    // Non-temporal b128 stores: output never re-read by this kernel.
#pragma unroll
    for (int t = 0; t < T_STEPS; ++t) {
        __builtin_nontemporal_store(ov[t], out + ((unsigned)t * perT4 + idx));
    }
}

extern "C" void kernel_launch(void* const* d_in, const int* in_sizes, int n_in,
                              void* d_out, int out_size, void* d_ws, size_t ws_size,
                              hipStream_t stream) {
    const f4* x = (const f4*)d_in[0];
    f4* out = (f4*)d_out;

    const long long total = (long long)in_sizes[0];      // T*B*N
    const unsigned perT4  = (unsigned)(total / T_STEPS / 4);  // (B*N)/4

    const int block = 256;
    const int grid = (int)((perT4 + block - 1) / block); // 8192 for B=2048,N=4096

    lif_fwd_kernel<<<grid, block, 0, stream>>>(x, out, perT4);
}